// Experts_75393855914558
// MI455X (gfx1250) — compile-verified
//
#include <hip/hip_runtime.h>
#include <hip/hip_bf16.h>

#define N_EXPERTS 8

// 16-byte int vector type matching the builtin's expected pointee
typedef int v4i __attribute__((vector_size(16)));

// ---------------------------------------------------------------------------
// CDNA5 async global->LDS copy (gfx1250).  Builtin signature (probe-confirmed):
// (v4i AS1* gaddr, v4i AS3* lds, imm off, imm cpol) ->
// GLOBAL_LOAD_ASYNC_TO_LDS_B128, tracked with ASYNCcnt
// (CDNA5 ISA).
// ---------------------------------------------------------------------------
__device__ __forceinline__ void async_load_b128_to_lds(const float* g, void* l) {
#if __has_builtin(__builtin_amdgcn_global_load_async_to_lds_b128)
    __builtin_amdgcn_global_load_async_to_lds_b128(
        (__attribute__((address_space(1))) v4i*)(g),
        (__attribute__((address_space(3))) v4i*)(l),
        0, 0);
#else
    unsigned lofs = (unsigned)(unsigned long long)l;
    asm volatile("global_load_async_to_lds_b128 %0, %1, off"
                 :: "v"(lofs), "v"(g) : "memory");
#endif
}

template <int N>
__device__ __forceinline__ void wait_asynccnt() {
    static_assert(N == 0 || N == 4, "only literals used below");
#if __has_builtin(__builtin_amdgcn_s_wait_asynccnt)
    __builtin_amdgcn_s_wait_asynccnt(N);
#else
    if constexpr (N == 0) asm volatile("s_wait_asynccnt 0" ::: "memory");
    else                  asm volatile("s_wait_asynccnt 4" ::: "memory");
#endif
    asm volatile("" ::: "memory");   // compiler-level fence: keep LDS reads below
}

// ---------------------------------------------------------------------------
// Heavy phase (FIRST in file so the disasm snippet shows it):
// out[row, :] = scale[row] * x[row, :] + bias
// 1024 blocks x 16 rows, double-buffered async global->LDS staging.
// Each wave async-loads exactly the 512 B its own lanes consume per row, so a
// per-wave s_wait_asynccnt is the full synchronization (no barriers needed).
// ---------------------------------------------------------------------------
__global__ void __launch_bounds__(256)
k_scale_bias_async(const float* __restrict__ x, const float* __restrict__ scale,
                   const float* __restrict__ bias, float* __restrict__ out) {
    constexpr int D = 1024, S = 4, R = 16, NST = R / S;
    __shared__ float lbuf[2][S][D];                    // 32 KB
    const int t = threadIdx.x;                         // 0..255
    const int col = t << 2;                            // 4 floats per thread
    const long long row0 = (long long)blockIdx.x * R;
    const float4 b4 = *reinterpret_cast<const float4*>(bias + col);

    // prologue: stage 0 -> buffer 0 (S async instructions per wave)
#pragma unroll
    for (int r = 0; r < S; ++r)
        async_load_b128_to_lds(x + (row0 + r) * D + col, &lbuf[0][r][col]);

    for (int s = 0; s < NST; ++s) {
        const int buf = s & 1;
        if (s + 1 < NST) {
#pragma unroll
            for (int r = 0; r < S; ++r)
                async_load_b128_to_lds(x + (row0 + (s + 1) * S + r) * D + col,
                                       &lbuf[buf ^ 1][r][col]);
            wait_asynccnt<S>();   // <= S outstanding: stage s fully landed in LDS
        } else {
            wait_asynccnt<0>();
        }
#pragma unroll
        for (int r = 0; r < S; ++r) {
            const long long row = row0 + s * S + r;
            const float sc = scale[row];
            const float4 v = *reinterpret_cast<const float4*>(&lbuf[buf][r][col]);
            float4 o;
            o.x = fmaf(sc, v.x, b4.x);
            o.y = fmaf(sc, v.y, b4.y);
            o.z = fmaf(sc, v.z, b4.z);
            o.w = fmaf(sc, v.w, b4.w);
            *reinterpret_cast<float4*>(out + row * D + col) = o;
        }
    }
}

// ---------------------------------------------------------------------------
// Routing phase: reproduce the stable-sort capacity drop exactly.
// Entry t survives iff |{t' < t : te[t'] == te[t]}| < C.
// ---------------------------------------------------------------------------
__global__ void k_zero_f32(float* p, int n) {
    int i = blockIdx.x * blockDim.x + threadIdx.x;
    if (i < n) p[i] = 0.0f;
}

// 512 entries per block -> per-block per-expert histogram
__global__ void k_count(const int* __restrict__ te, int* __restrict__ partial, int T) {
    __shared__ int cnt[N_EXPERTS];
    const int t = threadIdx.x;
    if (t < N_EXPERTS) cnt[t] = 0;
    __syncthreads();
    const int base = blockIdx.x * 512;
    const int i0 = base + 2 * t, i1 = i0 + 1;
    if (i0 < T) atomicAdd(&cnt[te[i0]], 1);
    if (i1 < T) atomicAdd(&cnt[te[i1]], 1);
    __syncthreads();
    if (t < N_EXPERTS) partial[blockIdx.x * N_EXPERTS + t] = cnt[t];
}

// offs[blk][e] = sum over blocks < blk of partial[.][e]  (exclusive scan)
__global__ void k_offsets(const int* __restrict__ partial, int* __restrict__ offs, int NB) {
    const int total = NB * N_EXPERTS;
    for (int i = threadIdx.x; i < total; i += blockDim.x) {
        const int blk = i >> 3, e = i & 7;
        int s = 0;
        for (int b = 0; b < blk; ++b) s += partial[b * N_EXPERTS + e];
        offs[i] = s;
    }
}

// local rank inside block + block offset = global rank within expert.
// Both entries of a token are handled by the same thread -> the two
// atomicAdds to scale[token] are in program order -> deterministic result.
__global__ void k_apply(const int* __restrict__ te, const float* __restrict__ ew,
                        const int* __restrict__ offs, float* __restrict__ scale,
                        int T, int C, int topk) {
    __shared__ int eid[512];
    const int t = threadIdx.x;
    const int base = blockIdx.x * 512;
    const int i0 = base + 2 * t, i1 = i0 + 1;
    const int e0 = (i0 < T) ? te[i0] : -1;
    const int e1 = (i1 < T) ? te[i1] : -1;
    eid[2 * t]     = e0;
    eid[2 * t + 1] = e1;
    __syncthreads();
    int r0 = 0, r1 = 0;
    for (int j = 0; j < 2 * t; ++j) {
        const int e = eid[j];
        r0 += (e == e0);
        r1 += (e == e1);
    }
    r1 += (e0 == e1);
    if (e0 >= 0) {
        const int pos = offs[blockIdx.x * N_EXPERTS + e0] + r0;
        if (pos < C) atomicAdd(&scale[i0 / topk], ew[i0]);
    }
    if (e1 >= 0) {
        const int pos = offs[blockIdx.x * N_EXPERTS + e1] + r1;
        if (pos < C) atomicAdd(&scale[i1 / topk], ew[i1]);
    }
}

// generic fallback (shape-robust, plain loads)
__global__ void k_scale_bias_generic(const float* __restrict__ x,
                                     const float* __restrict__ scale,
                                     const float* __restrict__ bias,
                                     float* __restrict__ out,
                                     long long total, int d) {
    long long i = (long long)blockIdx.x * blockDim.x + threadIdx.x;
    const long long stride = (long long)gridDim.x * blockDim.x;
    for (; i < total; i += stride)
        out[i] = fmaf(scale[i / d], x[i], bias[i % d]);
}

// ---------------------------------------------------------------------------
// inputs (setup_inputs order): 0:x 1:cond 2:mask 3:scores 4:expert_weights
//                              5:top_experts 6:bias   (cond/mask/scores unused)
// ---------------------------------------------------------------------------
extern "C" void kernel_launch(void* const* d_in, const int* in_sizes, int n_in,
                              void* d_out, int out_size, void* d_ws, size_t ws_size,
                              hipStream_t stream) {
    const float* x    = (const float*)d_in[0];
    const float* ew   = (const float*)d_in[4];
    const int*   te   = (const int*)d_in[5];
    const float* bias = (const float*)d_in[6];
    float*       out  = (float*)d_out;

    const int d      = in_sizes[6];                       // 1024
    const long long xN = in_sizes[0];                     // b*n*d
    const int rows   = (int)(xN / d);                     // 16384
    const int topk   = in_sizes[4] / rows;                // 2
    const int T      = rows * topk;                       // 32768
    const int C      = T / N_EXPERTS;                     // 4096
    const int NB     = (T + 511) / 512;                   // 64

    char* ws = (char*)d_ws;
    float* scale   = (float*)ws;                                            // rows f32
    int*   partial = (int*)(ws + (((size_t)rows * 4 + 255) & ~(size_t)255));// NB*8 i32
    int*   offs    = partial + (size_t)NB * N_EXPERTS;                      // NB*8 i32

    k_zero_f32<<<(rows + 255) / 256, 256, 0, stream>>>(scale, rows);
    k_count  <<<NB, 256, 0, stream>>>(te, partial, T);
    k_offsets<<<1, 512, 0, stream>>>(partial, offs, NB);
    k_apply  <<<NB, 256, 0, stream>>>(te, ew, offs, scale, T, C, topk);

    if (d == 1024 && rows % 16 == 0) {
        k_scale_bias_async<<<rows / 16, 256, 0, stream>>>(x, scale, bias, out);
    } else {
        const long long total = (long long)rows * d;
        int blocks = (int)((total + 255) / 256);
        if (blocks > 4096) blocks = 4096;
        k_scale_bias_generic<<<blocks, 256, 0, stream>>>(x, scale, bias, out, total, d);
    }
}